// Fact2seqAttn_66348654788979
// MI455X (gfx1250) — compile-verified
//
#include <hip/hip_runtime.h>

#define V_  50000
#define E_  1024
#define H_  1024
#define MCL 100
#define CL  100
#define FL  10
#define T_  32
#define SOS 1
#define K_  1024   // E_ == H_ == 1024, the shared GEMV/GEMM depth

typedef __attribute__((ext_vector_type(2))) float v2f;
typedef __attribute__((ext_vector_type(8))) float v8f;

// ---------------------------------------------------------------------------
// fe[c,:] = sum_f emb_in_z[context[c,f]] * pos[f,:]   (padding_idx=0 -> zero)
// ---------------------------------------------------------------------------
__global__ void fe_kernel(const int* __restrict__ context,
                          const float* __restrict__ emb_in,
                          float* __restrict__ fe) {
  int c  = blockIdx.x;
  int e0 = threadIdx.x * 4;
  float acc0 = 0.f, acc1 = 0.f, acc2 = 0.f, acc3 = 0.f;
  for (int f = 0; f < FL; ++f) {
    int tok = context[c * FL + f];
    if (tok != 0) {
      float s  = (float)f / (float)(FL - 1);
      float a  = 1.f - s;           // pos = (1-s) - e*(1-2s)
      float b  = 1.f - 2.f * s;
      const float* row = emb_in + (size_t)tok * E_;
      float4 rv = *(const float4*)(row + e0);
      float inv = 1.f / (float)(E_ - 1);
      acc0 += rv.x * (a - (float)(e0 + 0) * inv * b);
      acc1 += rv.y * (a - (float)(e0 + 1) * inv * b);
      acc2 += rv.z * (a - (float)(e0 + 2) * inv * b);
      acc3 += rv.w * (a - (float)(e0 + 3) * inv * b);
    }
  }
  float4 o = {acc0, acc1, acc2, acc3};
  *(float4*)(fe + (size_t)c * E_ + e0) = o;
}

// ---------------------------------------------------------------------------
// dec_emb[t,:] = emb_out_z[ t==0 ? SOS : description[t-1] ]
// ---------------------------------------------------------------------------
__global__ void dec_kernel(const int* __restrict__ description,
                           const float* __restrict__ emb_out,
                           float* __restrict__ dec) {
  int t  = blockIdx.x;
  int e0 = threadIdx.x * 4;
  int tok = (t == 0) ? SOS : description[t - 1];
  float4 v = {0.f, 0.f, 0.f, 0.f};
  if (tok != 0) v = *(const float4*)(emb_out + (size_t)tok * E_ + e0);
  *(float4*)(dec + (size_t)t * E_ + e0) = v;
}

// ---------------------------------------------------------------------------
// A_fe[c,m] = dot(fe[c,:], attn_w[m, 0:E])     (step-invariant attention term)
// ---------------------------------------------------------------------------
__global__ void afe_kernel(const float* __restrict__ fe,
                           const float* __restrict__ attn_w,
                           float* __restrict__ afe) {
  __shared__ float sfe[E_];
  int c = blockIdx.x;
  for (int i = threadIdx.x; i < E_; i += blockDim.x) sfe[i] = fe[(size_t)c * E_ + i];
  __syncthreads();
  int wave = threadIdx.x >> 5, lane = threadIdx.x & 31;
  for (int m = wave; m < MCL; m += 8) {
    const float* wr = attn_w + (size_t)m * (E_ + H_);
    float acc = 0.f;
    for (int j = 0; j < E_; j += 128) {
      float4 wv = *(const float4*)(wr + j + lane * 4);
      float4 xv = *(const float4*)(sfe + j + lane * 4);
      acc += wv.x * xv.x + wv.y * xv.y + wv.z * xv.z + wv.w * xv.w;
    }
    for (int off = 16; off > 0; off >>= 1) acc += __shfl_down(acc, off, 32);
    if (lane == 0) afe[(size_t)c * MCL + m] = acc;
  }
}

// ---------------------------------------------------------------------------
// Generic GEMV  y[i] = dot(W[i*ld ..], x) (+bias),  K = 1024 fixed
// ---------------------------------------------------------------------------
__global__ void gemv_kernel(const float* __restrict__ W, int ld,
                            const float* __restrict__ x,
                            const float* __restrict__ bias,
                            float* __restrict__ y, int N) {
  int wave = threadIdx.x >> 5, lane = threadIdx.x & 31;
  int row = blockIdx.x * 8 + wave;
  if (row >= N) return;
  const float* wr = W + (size_t)row * ld;
  float acc = 0.f;
  for (int j = 0; j < K_; j += 128) {
    float4 wv = *(const float4*)(wr + j + lane * 4);
    float4 xv = *(const float4*)(x + j + lane * 4);
    acc += wv.x * xv.x + wv.y * xv.y + wv.z * xv.z + wv.w * xv.w;
  }
  for (int off = 16; off > 0; off >>= 1) acc += __shfl_down(acc, off, 32);
  if (lane == 0) y[row] = bias ? (acc + bias[row]) : acc;
}

// y[i] = tanh(dot(W[i*ld ..], x) + add[i] + bias[i])   (fused out1 projection)
__global__ void gemv_tanh_kernel(const float* __restrict__ W, int ld,
                                 const float* __restrict__ x,
                                 const float* __restrict__ add,
                                 const float* __restrict__ bias,
                                 float* __restrict__ y, int N) {
  int wave = threadIdx.x >> 5, lane = threadIdx.x & 31;
  int row = blockIdx.x * 8 + wave;
  if (row >= N) return;
  const float* wr = W + (size_t)row * ld;
  float acc = 0.f;
  for (int j = 0; j < K_; j += 128) {
    float4 wv = *(const float4*)(wr + j + lane * 4);
    float4 xv = *(const float4*)(x + j + lane * 4);
    acc += wv.x * xv.x + wv.y * xv.y + wv.z * xv.z + wv.w * xv.w;
  }
  for (int off = 16; off > 0; off >>= 1) acc += __shfl_down(acc, off, 32);
  if (lane == 0) y[row] = tanhf(acc + add[row] + bias[row]);
}

// ---------------------------------------------------------------------------
// Attention tail: scores -> softmax -> ctx      (single block)
// ---------------------------------------------------------------------------
__global__ void attn_ctx_kernel(const float* __restrict__ afe,
                                const float* __restrict__ vh,
                                const float* __restrict__ attn_b,
                                const float* __restrict__ attn2_w,
                                const float* __restrict__ attn2_b,
                                const float* __restrict__ fe,
                                float* __restrict__ ctx) {
  __shared__ float sw[CL];
  __shared__ float sv[MCL];
  int tid = threadIdx.x, wave = tid >> 5, lane = tid & 31;
  if (tid < MCL) sv[tid] = vh[tid] + attn_b[tid];
  __syncthreads();
  for (int c = wave; c < CL; c += 8) {
    float acc = 0.f;
    for (int m = lane; m < MCL; m += 32)
      acc += tanhf(afe[(size_t)c * MCL + m] + sv[m]) * attn2_w[m];
    for (int off = 16; off > 0; off >>= 1) acc += __shfl_down(acc, off, 32);
    if (lane == 0) sw[c] = acc + attn2_b[0];
  }
  __syncthreads();
  if (wave == 0) {                      // softmax over CL=100, one wave
    float m = -1e30f;
    for (int u = 0; u < 4; ++u) { int c = lane + 32 * u; if (c < CL) m = fmaxf(m, sw[c]); }
    for (int off = 16; off > 0; off >>= 1) m = fmaxf(m, __shfl_xor(m, off, 32));
    float ex[4]; float s = 0.f;
    for (int u = 0; u < 4; ++u) {
      int c = lane + 32 * u;
      ex[u] = (c < CL) ? __expf(sw[c] - m) : 0.f;
      s += ex[u];
    }
    for (int off = 16; off > 0; off >>= 1) s += __shfl_xor(s, off, 32);
    float inv = 1.f / s;
    for (int u = 0; u < 4; ++u) { int c = lane + 32 * u; if (c < CL) sw[c] = ex[u] * inv; }
  }
  __syncthreads();
  int e0 = tid * 4;
  float4 acc = {0.f, 0.f, 0.f, 0.f};
  for (int c = 0; c < CL; ++c) {
    float w = sw[c];
    float4 fv = *(const float4*)(fe + (size_t)c * E_ + e0);
    acc.x += w * fv.x; acc.y += w * fv.y; acc.z += w * fv.z; acc.w += w * fv.w;
  }
  *(float4*)(ctx + e0) = acc;
}

// ---------------------------------------------------------------------------
// GRU gate combine (elementwise), h updated in place
// ---------------------------------------------------------------------------
__global__ void gru_kernel(const float* __restrict__ GXt,
                           const float* __restrict__ gxc,
                           const float* __restrict__ gh,
                           float* __restrict__ h) {
  int i0 = threadIdx.x * 4;
  for (int q = 0; q < 4; ++q) {
    int i = i0 + q;
    float rx = GXt[i]          + gxc[i];
    float zx = GXt[H_ + i]     + gxc[H_ + i];
    float nx = GXt[2 * H_ + i] + gxc[2 * H_ + i];
    float r = 1.f / (1.f + __expf(-(rx + gh[i])));
    float z = 1.f / (1.f + __expf(-(zx + gh[H_ + i])));
    float n = tanhf(nx + r * gh[2 * H_ + i]);
    h[i] = (1.f - z) * n + z * h[i];
  }
}

// ---------------------------------------------------------------------------
// WMMA GEMM:  out[32, N] = X[32, 1024] @ W[N, ld(>=1024)]^T  (+ bias[n])
// f32 path: V_WMMA_F32_16X16X4_F32. Each wave: one 16-col N-tile, both M-tiles
// (M=0..15 and 16..31) share the B (weight) stream.
// ---------------------------------------------------------------------------
__global__ void wmma_gemm_kernel(const float* __restrict__ X,
                                 const float* __restrict__ W, int ld,
                                 const float* __restrict__ bias,
                                 float* __restrict__ out, int N) {
  int wave  = threadIdx.x >> 5;
  int lane  = threadIdx.x & 31;
  int ntile = blockIdx.x * 8 + wave;
  if (ntile * 16 >= N) return;            // wave-uniform: EXEC stays all-1s
  int nbase = ntile * 16;
  int col16 = lane & 15;                  // N index in tile; M row index for A
  int khalf = (lane >> 4) * 2;            // K sub-offset 0 or 2
  const float* wrow = W + (size_t)(nbase + col16) * ld + khalf;   // B^T row
  const float* xr0  = X + (size_t)col16 * K_ + khalf;             // A, M-tile 0
  const float* xr1  = xr0 + 16 * K_;                              // A, M-tile 1
  v8f c0 = {}; v8f c1 = {};
#pragma unroll 4
  for (int kb = 0; kb < K_; kb += 4) {
    v2f b  = *(const v2f*)(wrow + kb);
    v2f a0 = *(const v2f*)(xr0 + kb);
    v2f a1 = *(const v2f*)(xr1 + kb);
    c0 = __builtin_amdgcn_wmma_f32_16x16x4_f32(false, a0, false, b, (short)0, c0, false, false);
    c1 = __builtin_amdgcn_wmma_f32_16x16x4_f32(false, a1, false, b, (short)0, c1, false, false);
  }
  float bv = bias ? bias[nbase + col16] : 0.f;
  int mhi = (lane >> 4) * 8;              // C/D layout: VGPR j -> M = j + mhi
#pragma unroll
  for (int j = 0; j < 8; ++j) {
    int m = j + mhi;
    out[(size_t)m * N + nbase + col16]        = c0[j] + bv;
    out[(size_t)(m + 16) * N + nbase + col16] = c1[j] + bv;
  }
}

// ---------------------------------------------------------------------------
// Per-row log-sum-exp + NLL over V=50000  (one block per t)
// ---------------------------------------------------------------------------
__global__ void lse_loss_kernel(const float* __restrict__ logits,
                                const int* __restrict__ description,
                                float* __restrict__ losses) {
  __shared__ float red[256];
  int t = blockIdx.x;
  const float* row = logits + (size_t)t * V_;
  float m = -1e30f;
  for (int i = threadIdx.x; i < V_; i += 256) m = fmaxf(m, row[i]);
  red[threadIdx.x] = m; __syncthreads();
  for (int s = 128; s > 0; s >>= 1) {
    if (threadIdx.x < s) red[threadIdx.x] = fmaxf(red[threadIdx.x], red[threadIdx.x + s]);
    __syncthreads();
  }
  m = red[0]; __syncthreads();
  float sum = 0.f;
  for (int i = threadIdx.x; i < V_; i += 256) sum += __expf(row[i] - m);
  red[threadIdx.x] = sum; __syncthreads();
  for (int s = 128; s > 0; s >>= 1) {
    if (threadIdx.x < s) red[threadIdx.x] += red[threadIdx.x + s];
    __syncthreads();
  }
  if (threadIdx.x == 0) {
    int y = description[t];
    losses[t] = -(row[y] - m - __logf(red[0]));
  }
}

__global__ void sum_kernel(const float* __restrict__ losses, float* __restrict__ out) {
  if (threadIdx.x == 0) {
    float s = 0.f;
    for (int t = 0; t < T_; ++t) s += losses[t];
    out[0] = s;
  }
}

// ---------------------------------------------------------------------------
extern "C" void kernel_launch(void* const* d_in, const int* in_sizes, int n_in,
                              void* d_out, int out_size, void* d_ws, size_t ws_size,
                              hipStream_t stream) {
  const int*   context     = (const int*)  d_in[0];
  const int*   description = (const int*)  d_in[1];
  const float* emb_in      = (const float*)d_in[2];
  const float* emb_out     = (const float*)d_in[3];
  const float* gru_w_ih    = (const float*)d_in[4];
  const float* gru_w_hh    = (const float*)d_in[5];
  const float* gru_b_ih    = (const float*)d_in[6];
  const float* gru_b_hh    = (const float*)d_in[7];
  const float* attn_w      = (const float*)d_in[8];
  const float* attn_b      = (const float*)d_in[9];
  const float* attn2_w     = (const float*)d_in[10];
  const float* attn2_b     = (const float*)d_in[11];
  const float* out1_w      = (const float*)d_in[12];
  const float* out1_b      = (const float*)d_in[13];
  const float* out2_w      = (const float*)d_in[14];
  const float* out2_b      = (const float*)d_in[15];

  float* ws = (float*)d_ws;
  float* fe     = ws;                    // 100*1024
  float* dec    = fe     + 102400;       // 32*1024
  float* afe    = dec    + 32768;        // 100*100 (pad)
  float* GX     = afe    + 10240;        // 32*3072
  float* h      = GX     + 98304;        // 1024
  float* vh     = h      + 1024;         // 100 (pad 256)
  float* gh     = vh     + 256;          // 3072
  float* ctx    = gh     + 3072;         // 1024
  float* gxc    = ctx    + 1024;         // 3072
  float* ucc    = gxc    + 3072;         // 1024
  float* O      = ucc    + 1024;         // 32*1024
  float* losses = O      + 32768;        // 32 (pad 64)
  float* logits = losses + 64;           // 32*50000

  // ---- batchable precompute ----
  fe_kernel <<<CL, 256, 0, stream>>>(context, emb_in, fe);
  dec_kernel<<<T_, 256, 0, stream>>>(description, emb_out, dec);
  afe_kernel<<<CL, 256, 0, stream>>>(fe, attn_w, afe);
  // GX[32,3072] = dec @ W_x^T + b_ih   (W_x = gru_w_ih[:, :1024], ld=2048)
  wmma_gemm_kernel<<<24, 256, 0, stream>>>(dec, gru_w_ih, 2 * E_, gru_b_ih, GX, 3 * H_);
  hipMemsetAsync(h, 0, H_ * sizeof(float), stream);

  // ---- sequential decoder: small L2-resident GEMVs per step ----
  for (int t = 0; t < T_; ++t) {
    gemv_kernel<<<13, 256, 0, stream>>>(attn_w + E_, E_ + H_, h, nullptr, vh, MCL);
    gemv_kernel<<<384, 256, 0, stream>>>(gru_w_hh, H_, h, gru_b_hh, gh, 3 * H_);
    attn_ctx_kernel<<<1, 256, 0, stream>>>(afe, vh, attn_b, attn2_w, attn2_b, fe, ctx);
    gemv_kernel<<<384, 256, 0, stream>>>(gru_w_ih + E_, 2 * E_, ctx, nullptr, gxc, 3 * H_);
    gemv_kernel<<<128, 256, 0, stream>>>(out1_w + H_, E_ + H_, ctx, nullptr, ucc, H_);
    gru_kernel<<<1, 256, 0, stream>>>(GX + (size_t)t * 3 * H_, gxc, gh, h);
    gemv_tanh_kernel<<<128, 256, 0, stream>>>(out1_w, E_ + H_, h, ucc, out1_b,
                                              O + (size_t)t * H_, H_);
  }

  // ---- the big one: logits[32,50000] = O @ out2_w^T + b, one 205MB pass ----
  wmma_gemm_kernel<<<391, 256, 0, stream>>>(O, out2_w, H_, out2_b, logits, V_);
  lse_loss_kernel<<<T_, 256, 0, stream>>>(logits, description, losses);
  sum_kernel<<<1, 32, 0, stream>>>(losses, (float*)d_out);
}